// Net_72043781423701
// MI455X (gfx1250) — compile-verified
//
#include <hip/hip_runtime.h>
#include <hip/hip_bf16.h>

// Problem constants (match reference)
#define NN      4096
#define F_IN    512
#define E0C     12288
#define EC      (E0C + NN)   // 16384 edges incl. self loops
#define CC      16
#define HID1    32
#define HID2    16
#define LPA_ITERS 5

typedef __attribute__((ext_vector_type(2))) float v2f;
typedef __attribute__((ext_vector_type(8))) float v8f;

// ---------------------------------------------------------------------------
// Edge decoding: edge_index is [2, E0] flat; self loops appended logically.
// ---------------------------------------------------------------------------
__device__ __forceinline__ void edge_rc(const int* __restrict__ ei, int e, int& r, int& c) {
    if (e < E0C) { r = ei[e]; c = ei[E0C + e]; }
    else         { r = e - E0C; c = e - E0C; }
}

// ---------------------------------------------------------------------------
// Kernel 0: ew0[row] = sum_j W_edge[row][j] + b_edge[row]
// 1 GiB streaming read -> THE bandwidth-critical kernel. One block per row,
// coalesced float4 (b128) loads, LDS tree reduction, prefetch next chunk.
// ---------------------------------------------------------------------------
__global__ void rowsum_kernel(const float* __restrict__ W, const float* __restrict__ b,
                              float* __restrict__ out) {
    __shared__ float red[256];
    const int row = blockIdx.x;
    const float4* __restrict__ Wr = (const float4*)(W + (size_t)row * EC);
    float s = 0.f;
    const int nvec = EC / 4;                       // 4096 float4 per row
    for (int i = threadIdx.x; i < nvec; i += 256) {
        __builtin_prefetch(Wr + i + 256, 0, 0);    // global_prefetch_b8
        float4 v = Wr[i];
        s += (v.x + v.y) + (v.z + v.w);
    }
    red[threadIdx.x] = s;
    __syncthreads();
    for (int off = 128; off > 0; off >>= 1) {
        if (threadIdx.x < (unsigned)off) red[threadIdx.x] += red[threadIdx.x + off];
        __syncthreads();
    }
    if (threadIdx.x == 0) out[row] = red[0] + b[row];
}

// ---------------------------------------------------------------------------
// Elementwise helpers
// ---------------------------------------------------------------------------
__global__ void zero_kernel(float* __restrict__ p, int n) {
    for (int i = blockIdx.x * blockDim.x + threadIdx.x; i < n; i += gridDim.x * blockDim.x)
        p[i] = 0.f;
}

__global__ void relu_kernel(float* __restrict__ p, int n) {
    for (int i = blockIdx.x * blockDim.x + threadIdx.x; i < n; i += gridDim.x * blockDim.x)
        p[i] = fmaxf(p[i], 0.f);
}

// deg[col[e]] += ew0[e]
__global__ void deg_scatter_kernel(const float* __restrict__ ew0, const int* __restrict__ ei,
                                   float* __restrict__ deg) {
    int e = blockIdx.x * blockDim.x + threadIdx.x;
    if (e >= EC) return;
    int r, c; edge_rc(ei, e, r, c);
    atomicAdd(&deg[c], ew0[e]);
}

// deg = isinf(deg) ? 0 : deg
__global__ void deg_fix_kernel(float* __restrict__ deg) {
    int i = blockIdx.x * blockDim.x + threadIdx.x;
    if (i >= NN) return;
    float d = deg[i];
    deg[i] = __builtin_isinf(d) ? 0.f : d;
}

// ew[e] = deg[row[e]] * ew0[e]
__global__ void ew_final_kernel(const float* __restrict__ ew0, const float* __restrict__ deg,
                                const int* __restrict__ ei, float* __restrict__ ew) {
    int e = blockIdx.x * blockDim.x + threadIdx.x;
    if (e >= EC) return;
    int r, c; edge_rc(ei, e, r, c);
    ew[e] = deg[r] * ew0[e];
}

// ---------------------------------------------------------------------------
// Dense GEMM via V_WMMA_F32_16X16X4_F32:  out[N, Nout] = A[N, K] @ W[K, Nout] + bias
// One wave (32 lanes) owns one 16x16 output tile. f32 fragment layout:
//   A 16x4 : lane%16 = M row; VGPR i holds K = 2*(lane/16) + i
//   B 4x16 : lane%16 = N col; VGPR i holds K = 2*(lane/16) + i
//   C/D    : lane%16 = N col; VGPR r holds M row = r + 8*(lane/16)
// Block = 128 threads = 4 waves, each with its own tile; no divergence so
// EXEC is all ones as WMMA requires.
// ---------------------------------------------------------------------------
__global__ void gemm_wmma_f32(const float* __restrict__ A, const float* __restrict__ W,
                              const float* __restrict__ bias, float* __restrict__ out,
                              int K, int Nout) {
    const int lane = threadIdx.x & 31;
    const int wave = threadIdx.x >> 5;
    const int ntiles = Nout >> 4;
    const int tile = blockIdx.x * (blockDim.x >> 5) + wave;
    const int mt = tile / ntiles;
    const int nt = tile % ntiles;
    const int h  = lane >> 4;     // half-wave: 0 or 1
    const int ln = lane & 15;     // M row (for A) / N col (for B, C, D)

    // bias broadcast: D[m][n] starts at bias[n] for all m
    const float bn = bias[nt * 16 + ln];
    v8f acc;
#pragma unroll
    for (int r = 0; r < 8; ++r) acc[r] = bn;

    const float* __restrict__ arow = A + (size_t)(mt * 16 + ln) * K;
    const float* __restrict__ wcol = W + (size_t)(nt * 16 + ln);
    for (int k0 = 0; k0 < K; k0 += 4) {
        v2f a, b;
        a.x = arow[k0 + 2 * h + 0];
        a.y = arow[k0 + 2 * h + 1];
        b.x = wcol[(size_t)(k0 + 2 * h + 0) * Nout];
        b.y = wcol[(size_t)(k0 + 2 * h + 1) * Nout];
        acc = __builtin_amdgcn_wmma_f32_16x16x4_f32(
            /*neg_a=*/false, a, /*neg_b=*/false, b,
            /*c_mod=*/(short)0, acc, /*reuse_a=*/false, /*reuse_b=*/false);
    }
#pragma unroll
    for (int r = 0; r < 8; ++r) {
        int m = r + 8 * h;
        out[(size_t)(mt * 16 + m) * Nout + nt * 16 + ln] = acc[r];
    }
}

// ---------------------------------------------------------------------------
// Edge message scatter: out[col[e], f] += ew[e] * src[row[e], f]
// ---------------------------------------------------------------------------
__global__ void scatter_kernel(const float* __restrict__ src, const float* __restrict__ ew,
                               const int* __restrict__ ei, float* __restrict__ out, int F) {
    int total = EC * F;
    for (int i = blockIdx.x * blockDim.x + threadIdx.x; i < total; i += gridDim.x * blockDim.x) {
        int e = i / F, f = i - e * F;
        int r, c; edge_rc(ei, e, r, c);
        float v = ew[e] * src[(size_t)r * F + f];
        atomicAdd(&out[(size_t)c * F + f], v);
    }
}

// ---------------------------------------------------------------------------
// LPA init: one-hot(y) * train_mask
// ---------------------------------------------------------------------------
__global__ void lpa_init_kernel(const int* __restrict__ y, const int* __restrict__ mask,
                                float* __restrict__ yh) {
    int i = blockIdx.x * blockDim.x + threadIdx.x;
    if (i >= NN * CC) return;
    int n = i / CC, c = i - n * CC;
    yh[i] = (y[n] == c && mask[n] != 0) ? 1.f : 0.f;
}

// ---------------------------------------------------------------------------
// Row-wise log_softmax over F=16 columns; one thread per row.
// ---------------------------------------------------------------------------
__global__ void logsoftmax_kernel(const float* __restrict__ in, float* __restrict__ out) {
    int i = blockIdx.x * blockDim.x + threadIdx.x;
    if (i >= NN) return;
    const float* r = in + (size_t)i * CC;
    float m = r[0];
#pragma unroll
    for (int f = 1; f < CC; ++f) m = fmaxf(m, r[f]);
    float s = 0.f;
#pragma unroll
    for (int f = 0; f < CC; ++f) s += expf(r[f] - m);
    float lse = m + logf(s);
#pragma unroll
    for (int f = 0; f < CC; ++f) out[(size_t)i * CC + f] = r[f] - lse;
}

// ---------------------------------------------------------------------------
// Launch sequence
// ---------------------------------------------------------------------------
extern "C" void kernel_launch(void* const* d_in, const int* in_sizes, int n_in,
                              void* d_out, int out_size, void* d_ws, size_t ws_size,
                              hipStream_t stream) {
    const float* x      = (const float*)d_in[0];
    const int*   ei     = (const int*)  d_in[1];
    const int*   y      = (const int*)  d_in[2];
    const int*   mask   = (const int*)  d_in[3];
    const float* W_edge = (const float*)d_in[4];
    const float* b_edge = (const float*)d_in[5];
    const float* W1 = (const float*)d_in[6];  const float* b1 = (const float*)d_in[7];
    const float* W2 = (const float*)d_in[8];  const float* b2 = (const float*)d_in[9];
    const float* W3 = (const float*)d_in[10]; const float* b3 = (const float*)d_in[11];
    const float* W4 = (const float*)d_in[12]; const float* b4 = (const float*)d_in[13];
    float* out = (float*)d_out;

    // Workspace layout (floats)
    float* ws  = (float*)d_ws;
    float* ew0 = ws;                 // E
    float* ew  = ew0 + EC;           // E
    float* deg = ew  + EC;           // N
    float* t   = deg + NN;           // N*32 (GEMM result scratch, max width)
    float* h1  = t   + NN * HID1;    // N*32
    float* h2  = h1  + NN * HID1;    // N*16
    float* h3  = h2  + NN * HID2;    // N*16
    float* h4  = h3  + NN * HID2;    // N*16 (logits)
    float* yhA = h4  + NN * CC;      // N*16
    float* yhB = yhA + NN * CC;      // N*16

    // 1) ew0 = rowsum(W_edge) + b_edge   -- the 1 GiB streaming kernel
    rowsum_kernel<<<EC, 256, 0, stream>>>(W_edge, b_edge, ew0);

    // 2) degree normalization
    zero_kernel<<<(NN + 255) / 256, 256, 0, stream>>>(deg, NN);
    deg_scatter_kernel<<<EC / 256, 256, 0, stream>>>(ew0, ei, deg);
    deg_fix_kernel<<<NN / 256, 256, 0, stream>>>(deg);
    ew_final_kernel<<<EC / 256, 256, 0, stream>>>(ew0, deg, ei, ew);

    // GCN layer 1: t = x@W1+b1 (WMMA), h1 = relu(scatter)
    gemm_wmma_f32<<<(NN / 16) * (HID1 / 16) / 4, 128, 0, stream>>>(x, W1, b1, t, F_IN, HID1);
    zero_kernel<<<(NN * HID1 + 255) / 256, 256, 0, stream>>>(h1, NN * HID1);
    scatter_kernel<<<(EC * HID1 + 255) / 256, 256, 0, stream>>>(t, ew, ei, h1, HID1);
    relu_kernel<<<(NN * HID1 + 255) / 256, 256, 0, stream>>>(h1, NN * HID1);

    // GCN layer 2
    gemm_wmma_f32<<<(NN / 16) * (HID2 / 16) / 4, 128, 0, stream>>>(h1, W2, b2, t, HID1, HID2);
    zero_kernel<<<(NN * HID2 + 255) / 256, 256, 0, stream>>>(h2, NN * HID2);
    scatter_kernel<<<(EC * HID2 + 255) / 256, 256, 0, stream>>>(t, ew, ei, h2, HID2);
    relu_kernel<<<(NN * HID2 + 255) / 256, 256, 0, stream>>>(h2, NN * HID2);

    // GCN layer 3
    gemm_wmma_f32<<<(NN / 16) * (HID2 / 16) / 4, 128, 0, stream>>>(h2, W3, b3, t, HID2, HID2);
    zero_kernel<<<(NN * HID2 + 255) / 256, 256, 0, stream>>>(h3, NN * HID2);
    scatter_kernel<<<(EC * HID2 + 255) / 256, 256, 0, stream>>>(t, ew, ei, h3, HID2);
    relu_kernel<<<(NN * HID2 + 255) / 256, 256, 0, stream>>>(h3, NN * HID2);

    // GCN layer 4 (no relu) -> logits h4
    gemm_wmma_f32<<<(NN / 16) * (CC / 16) / 4, 128, 0, stream>>>(h3, W4, b4, t, HID2, CC);
    zero_kernel<<<(NN * CC + 255) / 256, 256, 0, stream>>>(h4, NN * CC);
    scatter_kernel<<<(EC * CC + 255) / 256, 256, 0, stream>>>(t, ew, ei, h4, CC);

    // LPA: 5 propagation steps, ping-pong yhA/yhB (final lands in yhB)
    lpa_init_kernel<<<(NN * CC + 255) / 256, 256, 0, stream>>>(y, mask, yhA);
    float* src = yhA; float* dst = yhB;
    for (int it = 0; it < LPA_ITERS; ++it) {
        zero_kernel<<<(NN * CC + 255) / 256, 256, 0, stream>>>(dst, NN * CC);
        scatter_kernel<<<(EC * CC + 255) / 256, 256, 0, stream>>>(src, ew, ei, dst, CC);
        float* tmp = src; src = dst; dst = tmp;
    }
    // after loop, `src` points at the last-written buffer

    // Outputs: [log_softmax(h4); log_softmax(yh)]
    logsoftmax_kernel<<<(NN + 255) / 256, 256, 0, stream>>>(h4, out);
    logsoftmax_kernel<<<(NN + 255) / 256, 256, 0, stream>>>(src, out + (size_t)NN * CC);
}